// QuantumKernelMethod_65481071400555
// MI455X (gfx1250) — compile-verified
//
#include <hip/hip_runtime.h>

// Quantum kernel method, NQ=8 (DIM=256), DEPTH=2, B=16384.
// overlap = a(x)^dag a(y),  a(v) = R(v) * (D1 .* (P r(v)))
//   - final RZ layer + final CNOT permutation cancel inside |<sx|sy>|^2
//   - R(v) = A (x) B  ->  a = A * W * B^T : two 16x16x16 f32 GEMMs per
//     (state, re/im) on v_wmma_f32_16x16x4_f32 (K chunked by 4)
//   - CNOT permutation p() is GF(2)-linear: p(16m+k) = p(16m) ^ p(k), so the
//     product state r[p(b)] factorizes into a per-lane constant RH times
//     half-/compile-time-selected factors (no per-entry bit-product, no perm LDS).

#define NQ 8
#define DIM 256

typedef float v2f __attribute__((ext_vector_type(2)));
typedef float v8f __attribute__((ext_vector_type(8)));

// Relayout a 16x16 f32 matrix from WMMA C/D layout (row m: half=m>>3, vgpr=m&7)
// to B-operand layout (row k: half=(k>>1)&1, vgpr=2*(k>>2)+(k&1)).
// Only cross-lane move needed is lane L <-> L^16.
__device__ __forceinline__ v8f d_to_bop(v8f z, bool low) {
  float ex[8];
#pragma unroll
  for (int j = 0; j < 8; ++j) ex[j] = __shfl_xor((float)z[j], 16, 32);
  v8f b;
  b[0] = low ? z[0] : ex[2];
  b[1] = low ? z[1] : ex[3];
  b[2] = low ? z[4] : ex[6];
  b[3] = low ? z[5] : ex[7];
  b[4] = low ? ex[0] : z[2];
  b[5] = low ? ex[1] : z[3];
  b[6] = low ? ex[4] : z[6];
  b[7] = low ? ex[5] : z[7];
  return b;
}

struct LaneCtx {
  bool hb;            // lane >= 16 (K sub-offset bit: k bit1)
  bool mb[4];         // bits of m = lane & 15
  bool x01, x12, x23; // m0^m1, m1^m2, m2^m3 (high-nibble part of perm)
};

// Build a(v) (C/D layout) for one input vector. ph[kk][j2] = (cos,sin) of the
// RZ-layer-0 phase at basis index b = 16*m + (4*kk + 2*hb + j2), preloaded.
__device__ __forceinline__ void prep_state(const float* ang, const float2 ph[4][2],
                                           const LaneCtx& L, v8f& out_re, v8f& out_im) {
  float c[8], s[8], ns[8];
#pragma unroll
  for (int q = 0; q < 8; ++q) {
    float h = 0.5f * ang[q];
    s[q] = __sinf(h);
    c[q] = __cosf(h);
    ns[q] = -s[q];
  }

  // r[p(16m)] high-bit product: p(16m) = {m3, m2^m3, m1^m2, m0^m1 | bit3=m0}
  // bit t (4..7) <-> qubit 7-t
  float RH = (L.x01 ? s[3] : c[3]) * (L.x12 ? s[2] : c[2]);
  RH *= (L.x23 ? s[1] : c[1]);
  RH *= (L.mb[3] ? s[0] : c[0]);

  // perm low bits: bit3 = m0^kk1 (qubit4), bit2 = kk0^kk1 (qubit5, const pick),
  // bit1 = hb^kk0 (qubit6), bit0 = hb^j2 (qubit7)
  float hsel[2] = { L.mb[0] ? s[4] : c[4], L.mb[0] ? c[4] : s[4] };
  float dsel[2] = { L.hb ? s[6] : c[6], L.hb ? c[6] : s[6] };
  float esel[2] = { L.hb ? s[7] : c[7], L.hb ? c[7] : s[7] };

  // Ry factor tables: row bit rb runtime (bit t of m), col bit compile-time.
  // col=0: rb ? s : c ; col=1: rb ? c : -s     (Ry = [[c,-s],[s,c]])
  // A: qubits 0..3 (bit t <-> qubit 3-t) ; Bt: qubits 4..7 (bit t <-> qubit 7-t)
  float A0[4], A1[4], B0[4], B1[4];
#pragma unroll
  for (int t = 0; t < 4; ++t) {
    A0[t] = L.mb[t] ? s[3 - t] : c[3 - t];
    A1[t] = L.mb[t] ? c[3 - t] : ns[3 - t];
    B0[t] = L.mb[t] ? s[7 - t] : c[7 - t];
    B1[t] = L.mb[t] ? c[7 - t] : ns[7 - t];
  }
  float g1A = L.hb ? A1[1] : A0[1];
  float g1B = L.hb ? B1[1] : B0[1];

  // ---- stage 1: Z = W * B^T ----
  v8f zr = {0.f,0.f,0.f,0.f,0.f,0.f,0.f,0.f};
  v8f zi = {0.f,0.f,0.f,0.f,0.f,0.f,0.f,0.f};
#pragma unroll
  for (int kk = 0; kk < 4; ++kk) {
    const int kk0 = kk & 1, kk1 = kk >> 1;
    float base = RH * hsel[kk1];
    base *= (kk0 ^ kk1) ? s[5] : c[5];     // compile-time register pick
    base *= dsel[kk0];
    float wm0 = base * esel[0];
    float wm1 = base * esel[1];
    v2f wre = { wm0 * ph[kk][0].x, wm1 * ph[kk][1].x };
    v2f wim = { wm0 * ph[kk][0].y, wm1 * ph[kk][1].y };
    float pB = g1B * (kk0 ? B1[2] : B0[2]);
    pB *= (kk1 ? B1[3] : B0[3]);
    v2f bt = { B0[0] * pB, B1[0] * pB };
    zr = __builtin_amdgcn_wmma_f32_16x16x4_f32(false, wre, false, bt, (short)0, zr, false, false);
    zi = __builtin_amdgcn_wmma_f32_16x16x4_f32(false, wim, false, bt, (short)0, zi, false, false);
  }

  v8f zbr = d_to_bop(zr, !L.hb);
  v8f zbi = d_to_bop(zi, !L.hb);

  // ---- stage 2: a = A * Z ----
  v8f ar = {0.f,0.f,0.f,0.f,0.f,0.f,0.f,0.f};
  v8f ai = {0.f,0.f,0.f,0.f,0.f,0.f,0.f,0.f};
#pragma unroll
  for (int kk = 0; kk < 4; ++kk) {
    const int kk0 = kk & 1, kk1 = kk >> 1;
    float pA = g1A * (kk0 ? A1[2] : A0[2]);
    pA *= (kk1 ? A1[3] : A0[3]);
    v2f aop = { A0[0] * pA, A1[0] * pA };
    v2f br = { zbr[2 * kk], zbr[2 * kk + 1] };
    v2f bi = { zbi[2 * kk], zbi[2 * kk + 1] };
    ar = __builtin_amdgcn_wmma_f32_16x16x4_f32(false, aop, false, br, (short)0, ar, false, false);
    ai = __builtin_amdgcn_wmma_f32_16x16x4_f32(false, aop, false, bi, (short)0, ai, false, false);
  }
  out_re = ar;
  out_im = ai;
}

__global__ __launch_bounds__(256) void qkm_overlap_kernel(
    const float* __restrict__ x, const float* __restrict__ y,
    const float* __restrict__ params, float* __restrict__ out, int B) {
  __shared__ float2 s_ph[DIM];  // (cos,sin) of RZ-layer-0 phase per basis index

  {
    int b = threadIdx.x;  // blockDim.x == 256 == DIM
    float phi = 0.0f;
#pragma unroll
    for (int q = 0; q < 8; ++q)
      phi += (((b >> (7 - q)) & 1) ? 0.5f : -0.5f) * params[q];  // params row 0
    float sp, cp;
    sincosf(phi, &sp, &cp);
    s_ph[b] = make_float2(cp, sp);
  }
  __syncthreads();

  const int lane = threadIdx.x & 31;
  const int wave = threadIdx.x >> 5;
  const int e = blockIdx.x * 8 + wave;  // one wave per batch element
  if (e >= B) return;                   // wave-uniform: EXEC stays all-1s for WMMA

  LaneCtx L;
  const int m = lane & 15;
  L.hb = (lane & 16) != 0;
#pragma unroll
  for (int t = 0; t < 4; ++t) L.mb[t] = (m >> t) & 1;
  L.x01 = L.mb[0] ^ L.mb[1];
  L.x12 = L.mb[1] ^ L.mb[2];
  L.x23 = L.mb[2] ^ L.mb[3];

  // Phase factors for this lane's 8 basis indices (same for x and y states).
  float2 ph[4][2];
  const int kb = L.hb ? 2 : 0;
#pragma unroll
  for (int kk = 0; kk < 4; ++kk)
#pragma unroll
    for (int j2 = 0; j2 < 2; ++j2)
      ph[kk][j2] = s_ph[(m << 4) | (4 * kk + kb + j2)];

  float angx[8], angy[8];
#pragma unroll
  for (int q = 0; q < 8; ++q) {
    angx[q] = x[e * 8 + q];
    angy[q] = y[e * 8 + q];
  }

  v8f axr, axi, ayr, ayi;
  prep_state(angx, ph, L, axr, axi);
  prep_state(angy, ph, L, ayr, ayi);

  // overlap = sum conj(ax)*ay ; out = |overlap|^2
  float sr = 0.f, si = 0.f;
#pragma unroll
  for (int v = 0; v < 8; ++v) {
    sr += axr[v] * ayr[v] + axi[v] * ayi[v];
    si += axr[v] * ayi[v] - axi[v] * ayr[v];
  }
#pragma unroll
  for (int off = 16; off > 0; off >>= 1) {
    sr += __shfl_xor(sr, off, 32);
    si += __shfl_xor(si, off, 32);
  }
  if (lane == 0) out[e] = sr * sr + si * si;
}

extern "C" void kernel_launch(void* const* d_in, const int* in_sizes, int n_in,
                              void* d_out, int out_size, void* d_ws, size_t ws_size,
                              hipStream_t stream) {
  (void)n_in; (void)d_ws; (void)ws_size; (void)out_size;
  const float* x = (const float*)d_in[0];       // (B, 8) f32
  const float* y = (const float*)d_in[1];       // (B, 8) f32
  const float* params = (const float*)d_in[2];  // (2, 8) f32 (only row 0 survives algebra)
  float* out = (float*)d_out;                   // (B,) f32
  int B = in_sizes[0] / NQ;
  int blocks = (B + 7) / 8;                     // 8 waves (elements) per 256-thread block
  hipLaunchKernelGGL(qkm_overlap_kernel, dim3(blocks), dim3(256), 0, stream,
                     x, y, params, out, B);
}